// Discriminator_53077205844516
// MI455X (gfx1250) — compile-verified
//
#include <hip/hip_runtime.h>
#include <math.h>

typedef float v2f __attribute__((ext_vector_type(2)));
typedef float v4f __attribute__((ext_vector_type(4)));
typedef float v8f __attribute__((ext_vector_type(8)));

#define NN 4096
#define DD 128
#define TT 5
#define RR 4
#define GG 512
#define TP 8        // T padded to multiple of 4
#define HCATP 136   // T + D padded to multiple of 4

__device__ __forceinline__ v8f wmma_f32(v2f a, v2f b, v8f c) {
    // V_WMMA_F32_16X16X4_F32: exact fp32, D = A(16x4) * B(4x16) + C(16x16)
    return __builtin_amdgcn_wmma_f32_16x16x4_f32(false, a, false, b, (short)0, c,
                                                 false, false);
}

template <int ACT>
__device__ __forceinline__ float act_apply(float x) {
    if (ACT == 1) return tanhf(x);
    if (ACT == 2) return 1.0f / (1.0f + __expf(-x));
    return x;
}

// -----------------------------------------------------------------------------
// Guard-free fp32 WMMA GEMM: C[row,0:128] = act(A[row,0:K] @ W[0:K,0:128] + b)
// K MUST be a multiple of 4 (inputs are pre-padded with zeros in workspace).
// One wave -> 16 rows x 128 cols (8 v8f accumulators). blockDim=128 (4 waves)
// -> 64 rows/block. gridDim.y batches W (stride wbatch) and C (stride cbatch).
// -----------------------------------------------------------------------------
template <int ACT, bool HAS_BIAS>
__global__ void gemm128_kernel(const float* __restrict__ A, int lda, int K,
                               const float* __restrict__ W, long wbatch,
                               const float* __restrict__ bias,
                               float* __restrict__ C, long cbatch)
{
    const int b = blockIdx.y;
    W += (long)b * wbatch;
    C += (long)b * cbatch;

    const int wave = threadIdx.x >> 5;
    const int lane = threadIdx.x & 31;
    const int lo = lane & 15;
    const int hi = lane >> 4;               // 0 or 1
    const int row0 = blockIdx.x * 64 + wave * 16;
    const long arow = row0 + lo;

    v8f acc[8];
#pragma unroll
    for (int t = 0; t < 8; ++t)
#pragma unroll
        for (int v = 0; v < 8; ++v) acc[t][v] = 0.0f;

    const float* ap = A + arow * lda + 2 * hi;   // this half-wave's K pair
    for (int m = 0; m < K; m += 4) {
        const v2f aa = *reinterpret_cast<const v2f*>(ap + m);  // 1x global b64
        const float* w0 = W + (long)(m + 2 * hi) * DD + lo;
#pragma unroll
        for (int t = 0; t < 8; ++t) {
            v2f bb;
            bb.x = w0[t * 16];
            bb.y = w0[t * 16 + DD];
            acc[t] = wmma_f32(aa, bb, acc[t]);
        }
    }

#pragma unroll
    for (int t = 0; t < 8; ++t) {
        const int col = t * 16 + lo;
        const float bv = HAS_BIAS ? bias[col] : 0.0f;
#pragma unroll
        for (int v = 0; v < 8; ++v) {
            const int row = row0 + v + 8 * hi;   // C/D VGPR layout
            C[(long)row * DD + col] = act_apply<ACT>(acc[t][v] + bv);
        }
    }
}

// -----------------------------------------------------------------------------
// Heavy kernel: hout[n,e] = tanh(selfb[n,e] + sum_r sum_m a[r,n,m]*hrel[r,m,e])
// a: [R,N,N] streams from HBM exactly once; hrel K-panels (32x128) are staged
// in LDS in a (k-pair, col)-interleaved layout so each lane's B fragment is a
// single aligned ds_load_b64; row stride 288 words keeps the two half-waves on
// disjoint bank sets. Block = 4 waves = 64 rows sharing each panel.
// -----------------------------------------------------------------------------
#define KB 32
#define LSTR 288    // padded row stride (words) for a pair of k-rows
__global__ void rgcn_msg_kernel(const float* __restrict__ a,
                                const float* __restrict__ hrel,
                                const float* __restrict__ selfb,
                                float* __restrict__ hout)
{
    __shared__ float lds[(KB / 2) * LSTR];   // 18 KB

    const int wave = threadIdx.x >> 5;
    const int lane = threadIdx.x & 31;
    const int lo = lane & 15;
    const int hi = lane >> 4;
    const int row0 = blockIdx.x * 64 + wave * 16;
    const long arow = row0 + lo;
    const int tid = threadIdx.x;

    v8f acc[8];
#pragma unroll
    for (int t = 0; t < 8; ++t)
#pragma unroll
        for (int v = 0; v < 8; ++v) acc[t][v] = 0.0f;

    for (int r = 0; r < RR; ++r) {
        const float* aR = a    + (long)r * NN * NN;
        const float* hR = hrel + (long)r * NN * DD;
        for (int mb = 0; mb < NN; mb += KB) {
            __syncthreads();
            // stage hrel rows [mb, mb+KB): b128 global loads, repack to
            // paired layout lds[(k>>1)*LSTR + col*2 + (k&1)]
#pragma unroll
            for (int i = 0; i < (KB * DD) / (128 * 4); ++i) {
                const int base = i * 512 + tid * 4;      // 4 elems, same k-row
                const int k = base >> 7;
                const int col = base & 127;
                const v4f q = *reinterpret_cast<const v4f*>(
                    hR + (long)mb * DD + base);
                float* d = lds + (k >> 1) * LSTR + col * 2 + (k & 1);
                d[0] = q.x; d[2] = q.y; d[4] = q.z; d[6] = q.w;
            }
            __syncthreads();

            const float* ap = aR + arow * NN + mb + 2 * hi;
            __builtin_prefetch(ap + 2 * KB, 0, 0);       // a panel two ahead
#pragma unroll
            for (int m4 = 0; m4 < KB; m4 += 4) {
                const v2f aa = *reinterpret_cast<const v2f*>(ap + m4);
                const float* lrow = lds + (m4 / 2 + hi) * LSTR + lo * 2;
#pragma unroll
                for (int t = 0; t < 8; ++t) {
                    const v2f bb =
                        *reinterpret_cast<const v2f*>(lrow + t * 32);
                    acc[t] = wmma_f32(aa, bb, acc[t]);
                }
            }
        }
    }

#pragma unroll
    for (int t = 0; t < 8; ++t) {
        const int col = t * 16 + lo;
#pragma unroll
        for (int v = 0; v < 8; ++v) {
            const int row = row0 + v + 8 * hi;
            hout[(long)row * DD + col] =
                tanhf(selfb[(long)row * DD + col] + acc[t][v]);
        }
    }
}

// x0p[n, 0:5] = x0[n], x0p[n, 5:8] = 0
__global__ void padx_kernel(const float* __restrict__ x0, float* __restrict__ x0p)
{
    const int idx = blockIdx.x * blockDim.x + threadIdx.x;
    if (idx >= NN * TP) return;
    const int r = idx >> 3, c = idx & 7;
    x0p[idx] = (c < TT) ? x0[r * TT + c] : 0.0f;
}

// dst[b, 0:rows_src, :] = src[b], dst[b, rows_src:rows_dst, :] = 0
__global__ void padw_kernel(const float* __restrict__ src, float* __restrict__ dst,
                            int rows_src, int rows_dst)
{
    const long b = blockIdx.y;
    src += b * (long)rows_src * DD;
    dst += b * (long)rows_dst * DD;
    const int idx = blockIdx.x * blockDim.x + threadIdx.x;
    if (idx >= rows_dst * DD) return;
    dst[idx] = (idx < rows_src * DD) ? src[idx] : 0.0f;
}

// hcat[n, 0:5] = x0[n], hcat[n, 5:133] = h[n], hcat[n, 133:136] = 0
__global__ void hcat_kernel(const float* __restrict__ x0,
                            const float* __restrict__ h,
                            float* __restrict__ hcat)
{
    const int idx = blockIdx.x * blockDim.x + threadIdx.x;
    if (idx >= NN * HCATP) return;
    const int row = idx / HCATP;
    const int c = idx - row * HCATP;
    float v = 0.0f;
    if (c < TT) v = x0[row * TT + c];
    else if (c < TT + DD) v = h[row * DD + (c - TT)];
    hcat[idx] = v;
}

__global__ void mul_kernel(const float* __restrict__ a,
                           const float* __restrict__ b,
                           float* __restrict__ c, int n)
{
    const int idx = blockIdx.x * blockDim.x + threadIdx.x;
    if (idx < n) c[idx] = a[idx] * b[idx];
}

// pooled[g, col] = tanh(sum over rows n with seg[n]==g of garr[n, col])
__global__ void pool_kernel(const float* __restrict__ garr,
                            const int* __restrict__ seg,
                            float* __restrict__ pooled)
{
    const int g = blockIdx.x;
    const int col = threadIdx.x;
    int lo0 = 0, hi0 = NN;
    while (lo0 < hi0) { int mid = (lo0 + hi0) >> 1; if (seg[mid] < g) lo0 = mid + 1; else hi0 = mid; }
    int lo1 = lo0, hi1 = NN;
    while (lo1 < hi1) { int mid = (lo1 + hi1) >> 1; if (seg[mid] < g + 1) lo1 = mid + 1; else hi1 = mid; }
    float sum = 0.0f;
    for (int n = lo0; n < lo1; ++n) sum += garr[(long)n * DD + col];
    pooled[g * DD + col] = tanhf(sum);
}

// out[g] = tanh(pooled[g] @ fw1 + fb1) @ fw2 + fb2
__global__ void final_kernel(const float* __restrict__ pooled,
                             const float* __restrict__ fw1,
                             const float* __restrict__ fb1,
                             const float* __restrict__ fw2,
                             const float* __restrict__ fb2,
                             float* __restrict__ out)
{
    __shared__ float sp[DD];
    __shared__ float red[DD];
    const int g = blockIdx.x;
    const int tid = threadIdx.x;
    sp[tid] = pooled[g * DD + tid];
    __syncthreads();
    float t = fb1[tid];
    for (int e = 0; e < DD; ++e) t += sp[e] * fw1[e * DD + tid];
    t = tanhf(t);
    red[tid] = t * fw2[tid];
    __syncthreads();
    for (int s = 64; s > 0; s >>= 1) {
        if (tid < s) red[tid] += red[tid + s];
        __syncthreads();
    }
    if (tid == 0) out[g] = red[0] + fb2[0];
}

extern "C" void kernel_launch(void* const* d_in, const int* in_sizes, int n_in,
                              void* d_out, int out_size, void* d_ws, size_t ws_size,
                              hipStream_t stream)
{
    (void)in_sizes; (void)n_in; (void)out_size; (void)ws_size;

    const float* x0  = (const float*)d_in[0];   // [N,T]
    const float* a   = (const float*)d_in[1];   // [R,N,N]
    const int*   seg = (const int*)  d_in[2];   // [N]
    const float* w0s = (const float*)d_in[3];   // [T,D]
    const float* w0r = (const float*)d_in[4];   // [R,T,D]
    const float* b0  = (const float*)d_in[5];
    const float* w1s = (const float*)d_in[6];   // [D,D]
    const float* w1r = (const float*)d_in[7];   // [R,D,D]
    const float* b1  = (const float*)d_in[8];
    const float* iw1 = (const float*)d_in[9];   // [T+D,D]
    const float* ib1 = (const float*)d_in[10];
    const float* iw2 = (const float*)d_in[11];  // [D,D]
    const float* ib2 = (const float*)d_in[12];
    const float* jw1 = (const float*)d_in[13];
    const float* jb1 = (const float*)d_in[14];
    const float* jw2 = (const float*)d_in[15];
    const float* jb2 = (const float*)d_in[16];
    const float* fw1 = (const float*)d_in[17];
    const float* fb1 = (const float*)d_in[18];
    const float* fw2 = (const float*)d_in[19];  // [D,1]
    const float* fb2 = (const float*)d_in[20];
    float* out = (float*)d_out;                 // [G,1]

    // ---- workspace layout (floats); regions reused once dead ----
    float* ws    = (float*)d_ws;
    float* hrel  = ws;                                   // [R,N,D] = 2097152
    float* selfb = hrel + (long)RR * NN * DD;            // [N,D]
    float* h     = selfb + (long)NN * DD;                // [N,D]
    float* x0p   = h + (long)NN * DD;                    // [N,8]   = 32768
    float* w0sp  = x0p + (long)NN * TP;                  // [8,128]
    float* w0rp  = w0sp + TP * DD;                       // [R,8,128]
    float* iw1p  = w0rp + (long)RR * TP * DD;            // [136,128]
    float* jw1p  = iw1p + HCATP * DD;                    // [136,128]
    // reuse of the hrel region after the RGCN layers:
    float* hcat   = hrel;                                // [N,136]
    float* zi     = hrel + (long)NN * HCATP;             // [N,D]
    float* zj     = zi   + (long)NN * DD;                // [N,D]
    float* iout   = selfb;                               // [N,D] (selfb dead)
    float* jout   = h;                                   // [N,D] (h dead)
    float* garr   = hrel;                                // [N,D] (hcat dead)
    float* pooled = zj + (long)NN * DD;                  // [G,D]

    const dim3 blk(128);
    const dim3 grdR(NN / 64, RR);
    const dim3 grd1(NN / 64, 1);

    // ---- zero-pad inputs so every GEMM K is a multiple of 4 (no guards) ----
    padx_kernel<<<(NN * TP + 255) / 256, 256, 0, stream>>>(x0, x0p);
    padw_kernel<<<dim3((TP * DD + 255) / 256, RR), 256, 0, stream>>>(w0r, w0rp, TT, TP);
    padw_kernel<<<dim3((TP * DD + 255) / 256, 1), 256, 0, stream>>>(w0s, w0sp, TT, TP);
    padw_kernel<<<dim3((HCATP * DD + 255) / 256, 1), 256, 0, stream>>>(iw1, iw1p, TT + DD, HCATP);
    padw_kernel<<<dim3((HCATP * DD + 255) / 256, 1), 256, 0, stream>>>(jw1, jw1p, TT + DD, HCATP);

    // ---- RGCN layer 0 (K = 8) ----
    gemm128_kernel<0, false><<<grdR, blk, 0, stream>>>(x0p, TP, TP, w0rp, (long)TP * DD,
                                                       nullptr, hrel, (long)NN * DD);
    gemm128_kernel<0, true><<<grd1, blk, 0, stream>>>(x0p, TP, TP, w0sp, 0, b0, selfb, 0);
    rgcn_msg_kernel<<<NN / 64, blk, 0, stream>>>(a, hrel, selfb, h);

    // ---- RGCN layer 1 (K = 128) ----
    gemm128_kernel<0, false><<<grdR, blk, 0, stream>>>(h, DD, DD, w1r, (long)DD * DD,
                                                       nullptr, hrel, (long)NN * DD);
    gemm128_kernel<0, true><<<grd1, blk, 0, stream>>>(h, DD, DD, w1s, 0, b1, selfb, 0);
    rgcn_msg_kernel<<<NN / 64, blk, 0, stream>>>(a, hrel, selfb, h);  // h in-place ok

    // ---- concat [x0, h] (padded to 136 cols) ----
    hcat_kernel<<<(NN * HCATP + 255) / 256, 256, 0, stream>>>(x0, h, hcat);

    // ---- i-MLP / j-MLP (K = 136, zero-padded) ----
    gemm128_kernel<1, true><<<grd1, blk, 0, stream>>>(hcat, HCATP, HCATP, iw1p, 0, ib1, zi, 0);
    gemm128_kernel<1, true><<<grd1, blk, 0, stream>>>(hcat, HCATP, HCATP, jw1p, 0, jb1, zj, 0);
    gemm128_kernel<2, true><<<grd1, blk, 0, stream>>>(zi, DD, DD, iw2, 0, ib2, iout, 0); // sigmoid
    gemm128_kernel<1, true><<<grd1, blk, 0, stream>>>(zj, DD, DD, jw2, 0, jb2, jout, 0); // tanh

    // ---- gate, pool, final ----
    mul_kernel<<<(NN * DD + 255) / 256, 256, 0, stream>>>(iout, jout, garr, NN * DD);
    pool_kernel<<<GG, DD, 0, stream>>>(garr, seg, pooled);
    final_kernel<<<GG, DD, 0, stream>>>(pooled, fw1, fb1, fw2, fb2, out);
}